// BKTModel_73564199846001
// MI455X (gfx1250) — compile-verified
//
#include <hip/hip_runtime.h>
#include <math.h>
#include <stdint.h>

#define N_KCS 5000
#define BB    512
#define TT    100
#define TBL   (N_KCS * 5)   // 25000 floats = 100,000 bytes

// ---------------------------------------------------------------------------
// Kernel 1: probs_tbl = sigmoid(kc_logits), 25000 elements
// ---------------------------------------------------------------------------
__global__ void sigmoid_tbl_kernel(const float* __restrict__ logits,
                                   float* __restrict__ tbl) {
    int i = blockIdx.x * blockDim.x + threadIdx.x;
    if (i < TBL) {
        float x = logits[i];
        tbl[i] = 1.0f / (1.0f + expf(-x));
    }
}

// ---------------------------------------------------------------------------
// Kernel 2: state[b,k] = probs_tbl[k,4]  (broadcast init, 512*5000 floats).
// Vectorized: one float4 (global_store_b128) per thread; 5000 % 4 == 0 so a
// vector never straddles a batch row. Table gathers hit L0/L2 (100KB table,
// reused 512x). This is the only bandwidth-significant kernel (~10.3 MB).
// ---------------------------------------------------------------------------
__global__ void state_init_kernel(const float* __restrict__ tbl,
                                  float4* __restrict__ state4) {
    int i = blockIdx.x * blockDim.x + threadIdx.x;   // over (BB*N_KCS)/4
    if (i < (BB * N_KCS) / 4) {
        int k = (i * 4) % N_KCS;                     // row-local, multiple of 4
        float4 v;
        v.x = tbl[(k + 0) * 5 + 4];
        v.y = tbl[(k + 1) * 5 + 4];
        v.z = tbl[(k + 2) * 5 + 4];
        v.w = tbl[(k + 3) * 5 + 4];
        state4[i] = v;
    }
}

// ---------------------------------------------------------------------------
// Kernel 3: the BKT scan. One thread per batch chain (512 chains total).
// The 100KB sigmoid table is staged into LDS via async global->LDS DMA
// (CDNA5 ASYNCcnt path), then each step gathers from LDS and does the
// state read-modify-write through L2 (10 MB state << 192 MB L2).
// ---------------------------------------------------------------------------
__global__ __launch_bounds__(64) void bkt_scan_kernel(
    const int*   __restrict__ prev_kc,
    const int*   __restrict__ curr_kc,
    const int*   __restrict__ prev_corr,
    const float* __restrict__ tbl,
    float* __restrict__ probs,
    float* __restrict__ state)
{
    __shared__ float s_tbl[TBL];  // 100,000 bytes — fits in 320KB WGP LDS

    // ---- Stage table: 6250 x 16B chunks, async global->LDS ----
    {
        unsigned     lbase = (unsigned)(uintptr_t)(&s_tbl[0]);  // LDS byte offset
        const char*  gbase = (const char*)tbl;
        for (int i = threadIdx.x; i < TBL / 4; i += 64) {
            unsigned    laddr = lbase + 16u * (unsigned)i;
            const char* gaddr = gbase + 16u * (unsigned)i;
            asm volatile("global_load_async_to_lds_b128 %0, %1, off"
                         :: "v"(laddr), "v"(gaddr)
                         : "memory");
        }
        asm volatile("s_wait_asynccnt 0" ::: "memory");
    }
    __syncthreads();

    int b = blockIdx.x * 64 + threadIdx.x;
    if (b >= BB) return;

    const int* pk_row = prev_kc   + b * TT;
    const int* ck_row = curr_kc   + b * TT;
    const int* pc_row = prev_corr + b * TT;
    float*     st     = state + (size_t)b * N_KCS;
    float*     pr     = probs + (size_t)b * TT;

    // Pull the per-chain index rows toward the caches (gfx1250 prefetch)
    __builtin_prefetch(pk_row, 0, 0);
    __builtin_prefetch(ck_row, 0, 0);
    __builtin_prefetch(pc_row, 0, 0);

    // ---- t = 0: state is still the broadcast init, read p_init from LDS ----
    {
        int   ck0 = ck_row[0];
        float p2  = s_tbl[ck0 * 5 + 2];
        float p3  = s_tbl[ck0 * 5 + 3];
        float cs0 = s_tbl[ck0 * 5 + 4];     // == state0[b, ck0]
        pr[0] = p2 * (1.0f - cs0) + p3 * cs0;
    }

    // ---- t = 1 .. T-1: sequential recurrence ----
    for (int t = 1; t < TT; ++t) {
        int pk = pk_row[t];
        int ck = ck_row[t];
        int c  = pc_row[t];

        float pp0 = s_tbl[pk * 5 + 0];
        float pp1 = s_tbl[pk * 5 + 1];
        float pp2 = s_tbl[pk * 5 + 2];
        float pp3 = s_tbl[pk * 5 + 3];

        float skill = st[pk];

        // p_out = pch^c * (1-pch)^(1-c) with c in {0,1}  ->  select
        float po0 = c ? pp2 : 1.0f - pp2;
        float po1 = c ? pp3 : 1.0f - pp3;

        float num        = po1 * skill;
        float filt       = num / (po0 * (1.0f - skill) + num);
        float predictive = pp0 * (1.0f - filt) + (1.0f - pp1) * filt;

        st[pk] = predictive;

        // cs reads post-update state; shortcut the ck==pk dependent reload
        float cs = (ck == pk) ? predictive : st[ck];

        float cp2 = s_tbl[ck * 5 + 2];
        float cp3 = s_tbl[ck * 5 + 3];
        pr[t] = cp2 * (1.0f - cs) + cp3 * cs;
    }
}

// ---------------------------------------------------------------------------
// kernel_launch
// d_in: prev_kc (B*T i32), curr_kc (B*T i32), prev_corr (B*T i32),
//       kc_logits (N_KCS*5 f32)
// d_out: probs (B*T f32) ++ state_f (B*N_KCS f32)
// d_ws:  probs_tbl (N_KCS*5 f32)
// ---------------------------------------------------------------------------
extern "C" void kernel_launch(void* const* d_in, const int* in_sizes, int n_in,
                              void* d_out, int out_size, void* d_ws, size_t ws_size,
                              hipStream_t stream) {
    (void)in_sizes; (void)n_in; (void)out_size; (void)ws_size;

    const int*   prev_kc   = (const int*)d_in[0];
    const int*   curr_kc   = (const int*)d_in[1];
    const int*   prev_corr = (const int*)d_in[2];
    const float* kc_logits = (const float*)d_in[3];

    float* out   = (float*)d_out;
    float* probs = out;               // B*T
    float* state = out + BB * TT;     // B*N_KCS
    float* tbl   = (float*)d_ws;      // N_KCS*5

    sigmoid_tbl_kernel<<<(TBL + 255) / 256, 256, 0, stream>>>(kc_logits, tbl);

    state_init_kernel<<<((BB * N_KCS / 4) + 255) / 256, 256, 0, stream>>>(
        tbl, (float4*)state);

    bkt_scan_kernel<<<BB / 64, 64, 0, stream>>>(prev_kc, curr_kc, prev_corr,
                                                tbl, probs, state);
}